// UMAP_loss_68075231642296
// MI455X (gfx1250) — compile-verified
//
#include <hip/hip_runtime.h>

typedef __attribute__((ext_vector_type(2))) float v2f;
typedef __attribute__((ext_vector_type(8))) float v8f;

#define N_PTS   32768
#define TILE    16
#define ITILES  (N_PTS / TILE)        /* 2048 i-tiles */
#define WAVES_PB 8                    /* 256 threads = 8 wave32 */
#define RBLK_X  (ITILES / WAVES_PB)   /* 256 blocks in x */
#define JSPLIT  8                     /* split the j loop 8 ways for occupancy */
#define JT_PER  (ITILES / JSPLIT)     /* 256 j-tiles per block */
#define JCHUNK  (JT_PER * TILE)       /* 4096 points staged in LDS per block */
#define RB      (RBLK_X * JSPLIT)     /* 2048 repulsion partials */
#define ATT_BLOCKS 512
#define NPART   (RB + ATT_BLOCKS)

#define EPS_C   1e-4f
#define NSR_C   5.0f
#define LN2_C   0.69314718055994530942f
#define INV2N_C (1.0f / (2.0f * (float)N_PTS))

__device__ __forceinline__ float waveReduce(float v) {
#pragma unroll
    for (int off = 16; off > 0; off >>= 1) v += __shfl_down(v, off, 32);
    return v;
}

// ---------------- repulsive all-pairs term (WMMA 16x16x4 f32) ----------------
__global__ __launch_bounds__(256) void umap_repulse(const float* __restrict__ emb,
                                                    const float* __restrict__ deg,
                                                    float* __restrict__ partial) {
    __shared__ float sEx[JCHUNK];
    __shared__ float sEy[JCHUNK];
    __shared__ float sDg[JCHUNK];
    __shared__ float sRed[WAVES_PB];

    const int lane  = threadIdx.x & 31;
    const int wave  = threadIdx.x >> 5;
    const int laneN = lane & 15;
    const bool hi   = lane >= 16;

    // Stage this block's 4096-point j-chunk into LDS (x, y, degree).
    const int jBase = blockIdx.y * JCHUNK;
    for (int k = threadIdx.x; k < JCHUNK; k += blockDim.x) {
        const int g = jBase + k;
        sEx[k] = emb[2 * g + 0];
        sEy[k] = emb[2 * g + 1];
        sDg[k] = deg[g];
    }
    __syncthreads();

    // Per-wave i-tile state (invariant across the j loop).
    const int iTile = blockIdx.x * WAVES_PB + wave;  // 0..2047
    const int i0    = iTile * TILE;

    // A fragment: 16x4, A = -2*e_i. Lanes 0-15: K=0,1 in v[0],v[1]; lanes 16-31: K=2,3 = 0.
    const float ax = emb[2 * (i0 + laneN) + 0];
    const float ay = emb[2 * (i0 + laneN) + 1];
    v2f A;
    A.x = hi ? 0.0f : -2.0f * ax;
    A.y = hi ? 0.0f : -2.0f * ay;

    // Per-lane row norms and degrees for the 8 M-rows this lane's C/D slots cover.
    float cni[8], dgi[8];
    const int mbase = i0 + (hi ? 8 : 0);
#pragma unroll
    for (int v = 0; v < 8; ++v) {
        const float ex = emb[2 * (mbase + v) + 0];
        const float ey = emb[2 * (mbase + v) + 1];
        cni[v] = ex * ex + ey * ey;
        dgi[v] = deg[mbase + v];
    }

    float acc = 0.0f;  // sum of (deg_i + deg_j) * log2(inv_low_sim)
    for (int jt = 0; jt < JT_PER; ++jt) {
        const int jl  = jt * TILE + laneN;
        const float bx  = sEx[jl];
        const float by  = sEy[jl];
        const float nj  = bx * bx + by * by;
        const float dgj = sDg[jl];

        v2f B;
        B.x = hi ? 0.0f : bx;
        B.y = hi ? 0.0f : by;

        v8f C;
#pragma unroll
        for (int v = 0; v < 8; ++v) C[v] = cni[v] + nj;

        // D = (-2 e_i) . e_j + (|e_i|^2 + |e_j|^2) = squared distance tile (16x16)
        v8f Dm = __builtin_amdgcn_wmma_f32_16x16x4_f32(
            /*neg_a=*/false, A, /*neg_b=*/false, B,
            /*c_mod=*/(short)0, C, /*reuse_a=*/false, /*reuse_b=*/false);

#pragma unroll
        for (int v = 0; v < 8; ++v) {
            const float sqd = Dm[v];
            const float low = __builtin_amdgcn_rcpf(1.0f + sqd);      // v_rcp_f32
            const float s   = fmaxf(low - EPS_C, 0.0f);               // relu(low - eps)
            const float lg  = __builtin_amdgcn_logf(1.0f - s);        // v_log_f32 (log2)
            acc += (dgi[v] + dgj) * lg;
        }
    }

    acc = waveReduce(acc);
    if (lane == 0) sRed[wave] = acc;
    __syncthreads();
    if (threadIdx.x == 0) {
        float s = 0.0f;
#pragma unroll
        for (int w = 0; w < WAVES_PB; ++w) s += sRed[w];
        partial[blockIdx.y * RBLK_X + blockIdx.x] = s;
    }
}

// ---------------- attractive term over positive edges ----------------
__global__ __launch_bounds__(256) void umap_attract(const float* __restrict__ emb,
                                                    const float* __restrict__ sh,
                                                    const int* __restrict__ heads,
                                                    const int* __restrict__ tails,
                                                    int E, float* __restrict__ partial) {
    __shared__ float sRed[8];
    const int lane = threadIdx.x & 31;
    const int wave = threadIdx.x >> 5;

    float acc = 0.0f;  // sum of sh * log2(1 + sqd)
    for (int e = blockIdx.x * blockDim.x + threadIdx.x; e < E; e += gridDim.x * blockDim.x) {
        const int h = heads[e];
        const int t = tails[e];
        const float dx = emb[2 * h + 0] - emb[2 * t + 0];
        const float dy = emb[2 * h + 1] - emb[2 * t + 1];
        const float sqd = dx * dx + dy * dy;
        acc += sh[e] * __builtin_amdgcn_logf(1.0f + sqd);
    }

    acc = waveReduce(acc);
    if (lane == 0) sRed[wave] = acc;
    __syncthreads();
    if (threadIdx.x == 0) {
        float s = 0.0f;
#pragma unroll
        for (int w = 0; w < 8; ++w) s += sRed[w];
        partial[blockIdx.x] = s;
    }
}

// ---------------- deterministic final reduction ----------------
__global__ __launch_bounds__(256) void umap_final(const float* __restrict__ partial,
                                                  float* __restrict__ out) {
    __shared__ float sRed[8];
    const int lane = threadIdx.x & 31;
    const int wave = threadIdx.x >> 5;

    // loss_a = -ln2 * accA            -> contributes  +ln2 * accA to output
    // loss_r = NSR/(2N) * ln2 * accR  -> contributes  -NSR/(2N)*ln2 * accR
    const float CR = -NSR_C * INV2N_C * LN2_C;
    const float CA = LN2_C;

    float acc = 0.0f;
    for (int i = threadIdx.x; i < NPART; i += blockDim.x) {
        const float c = (i < RB) ? CR : CA;
        acc += c * partial[i];
    }

    acc = waveReduce(acc);
    if (lane == 0) sRed[wave] = acc;
    __syncthreads();
    if (threadIdx.x == 0) {
        float s = 0.0f;
#pragma unroll
        for (int w = 0; w < 8; ++w) s += sRed[w];
        out[0] = s;
    }
}

extern "C" void kernel_launch(void* const* d_in, const int* in_sizes, int n_in,
                              void* d_out, int out_size, void* d_ws, size_t ws_size,
                              hipStream_t stream) {
    (void)n_in; (void)out_size; (void)ws_size;
    const float* emb   = (const float*)d_in[0];
    const float* sh    = (const float*)d_in[1];
    const float* deg   = (const float*)d_in[2];
    const int*   heads = (const int*)d_in[3];
    const int*   tails = (const int*)d_in[4];
    const int E = in_sizes[3];

    float* partial = (float*)d_ws;   // [0,RB): repulsion, [RB,RB+ATT_BLOCKS): attraction

    dim3 rgrid(RBLK_X, JSPLIT);
    umap_repulse<<<rgrid, 256, 0, stream>>>(emb, deg, partial);
    umap_attract<<<ATT_BLOCKS, 256, 0, stream>>>(emb, sh, heads, tails, E, partial + RB);
    umap_final<<<1, 256, 0, stream>>>(partial, (float*)d_out);
}